// VQEmbedding_33277406609673
// MI455X (gfx1250) — compile-verified
//
#include <hip/hip_runtime.h>
#include <hip/hip_bf16.h>
#include <cstdint>

// Problem constants (match reference: K=4096 codes, D=512 dim, N=32768 vectors)
#define KDIM 4096
#define DDIM 512
#define NROW 32768

typedef __attribute__((ext_vector_type(2))) float v2f;
typedef __attribute__((ext_vector_type(8))) float v8f;

#define BM 128          // rows of Z per block
#define BN 128          // codebook entries per block
#define KC 32           // k-chunk staged through LDS
#define KP (KC / 2)     // float2 pairs per chunk
#define LDT 129         // padded LDS row stride (in float2) -> conflict-light

// D = Z * CB^T using V_WMMA_F32_16X16X4_F32 (native f32 matrix path on CDNA5).
__global__ __launch_bounds__(256)
void vq_logits_gemm(const float* __restrict__ Z, const float* __restrict__ CB,
                    float* __restrict__ out) {
  __shared__ float2 As[KP * LDT];   // As[kpair][row]  = Z[m0+row][2kp..2kp+1]
  __shared__ float2 Bs[KP * LDT];   // Bs[kpair][col]  = CB[n0+col][2kp..2kp+1]

  const int t    = threadIdx.x;
  const int lane = t & 31;
  const int wave = t >> 5;
  const int half = (lane >> 4) & 1;   // lane half selects K pair (ISA 16x4 layout)
  const int l    = lane & 15;         // M index (A) / N index (B,C,D)
  const int wm   = wave & 3;          // 4 waves along M  -> 32 rows each
  const int wn   = wave >> 2;         // 2 waves along N  -> 64 cols each
  const int m0   = blockIdx.y * BM;
  const int n0   = blockIdx.x * BN;

  v8f acc[2][4];
#pragma unroll
  for (int i = 0; i < 2; ++i)
#pragma unroll
    for (int j = 0; j < 4; ++j)
#pragma unroll
      for (int r = 0; r < 8; ++r) acc[i][j][r] = 0.0f;

  const int ktiles = DDIM / KC;   // 16
  for (int kt = 0; kt < ktiles; ++kt) {
    const int kb = kt * KC;
    // Stage A and B chunks: 128 rows x 16 float2 pairs each.
    // Consecutive threads read consecutive float2 within a row -> coalesced.
#pragma unroll
    for (int i = 0; i < 8; ++i) {
      const int idx = t + i * 256;
      const int kk  = idx & (KP - 1);   // pair index 0..15
      const int row = idx >> 4;         // 0..127
      As[kk * LDT + row] =
          *(const float2*)(Z + (size_t)(m0 + row) * DDIM + kb + 2 * kk);
      Bs[kk * LDT + row] =
          *(const float2*)(CB + (size_t)(n0 + row) * DDIM + kb + 2 * kk);
    }
    if (kt + 1 < ktiles) {
      const int kk  = t & (KP - 1);
      const int row = t >> 4;
      __builtin_prefetch(Z + (size_t)(m0 + row) * DDIM + kb + KC + 2 * kk, 0, 1);
      __builtin_prefetch(CB + (size_t)(n0 + row) * DDIM + kb + KC + 2 * kk, 0, 1);
    }
    __syncthreads();

    // 8 k-steps of 4 per chunk; ISA A layout: VGPR0=K{0,2}, VGPR1=K{1,3}
    // split across lane halves -> one float2 LDS read per fragment.
#pragma unroll
    for (int p2 = 0; p2 < KC / 4; ++p2) {
      const int pp = 2 * p2 + half;     // which float2 pair this lane-half needs
      v2f afr[2], bfr[4];
#pragma unroll
      for (int i = 0; i < 2; ++i)
        afr[i] = *(const v2f*)&As[pp * LDT + wm * 32 + i * 16 + l];
#pragma unroll
      for (int j = 0; j < 4; ++j)
        bfr[j] = *(const v2f*)&Bs[pp * LDT + wn * 64 + j * 16 + l];
#pragma unroll
      for (int i = 0; i < 2; ++i)
#pragma unroll
        for (int j = 0; j < 4; ++j)
          acc[i][j] = __builtin_amdgcn_wmma_f32_16x16x4_f32(
              /*neg_a=*/false, afr[i], /*neg_b=*/false, bfr[j],
              /*c_mod=*/(short)0, acc[i][j],
              /*reuse_a=*/false, /*reuse_b=*/false);
    }
    __syncthreads();
  }

  // C/D 16x16 f32 layout: VGPR r -> row r (lanes 0-15) / row r+8 (lanes 16-31).
  // Streaming 512MB output: nontemporal so the codebook stays resident in L2.
#pragma unroll
  for (int i = 0; i < 2; ++i) {
    const int rbase = m0 + wm * 32 + i * 16 + half * 8;
#pragma unroll
    for (int j = 0; j < 4; ++j) {
      const int cbase = n0 + wn * 64 + j * 16 + l;
#pragma unroll
      for (int r = 0; r < 8; ++r)
        __builtin_nontemporal_store(acc[i][j][r],
                                    out + (size_t)(rbase + r) * KDIM + cbase);
    }
  }
}

// One block per row: argmax over K=4096 logits (first-max tie-break like jnp).
__global__ __launch_bounds__(256)
void vq_argmax_rows(const float* __restrict__ logits, float* __restrict__ idx_out) {
  const int n = blockIdx.x;
  const float* row = logits + (size_t)n * KDIM;
  const int t = threadIdx.x;

  float best = -__builtin_inff();
  int bi = 0;
  for (int k = t; k < KDIM; k += 256) {
    const float v = __builtin_nontemporal_load(row + k);
    if (v > best) { best = v; bi = k; }
  }
  // wave32 reduction
#pragma unroll
  for (int off = 16; off > 0; off >>= 1) {
    const float ov = __shfl_down(best, off, 32);
    const int   oi = __shfl_down(bi, off, 32);
    if (ov > best || (ov == best && oi < bi)) { best = ov; bi = oi; }
  }
  __shared__ float sv[8];
  __shared__ int   si[8];
  if ((t & 31) == 0) { sv[t >> 5] = best; si[t >> 5] = bi; }
  __syncthreads();
  if (t == 0) {
#pragma unroll
    for (int w = 1; w < 8; ++w)
      if (sv[w] > best || (sv[w] == best && si[w] < bi)) { best = sv[w]; bi = si[w]; }
    idx_out[n] = (float)bi;
  }
}

extern "C" void kernel_launch(void* const* d_in, const int* in_sizes, int n_in,
                              void* d_out, int out_size, void* d_ws, size_t ws_size,
                              hipStream_t stream) {
  const float* z  = (const float*)d_in[0];   // [N, D] f32
  const float* cb = (const float*)d_in[1];   // [K, D] f32
  float* out = (float*)d_out;                // [N*K logits][N indices] flat

  dim3 grid(KDIM / BN, NROW / BM);           // 32 x 256 blocks
  vq_logits_gemm<<<grid, 256, 0, stream>>>(z, cb, out);

  float* idx_out = out + (size_t)NROW * KDIM;
  vq_argmax_rows<<<NROW, 256, 0, stream>>>(out, idx_out);
}